// ApexTransducerLoss_14061722927179
// MI455X (gfx1250) — compile-verified
//
#include <hip/hip_runtime.h>
#include <stdint.h>

// ---------------------------------------------------------------------------
// RNNT loss for MI455X (gfx1250, wave32).
// Stage 1 (HBM-bound, ~545MB -> ~23us roofline): per-wave logsumexp over the
//   1024-wide vocab axis; wave reduction done on the matrix pipe with
//   V_WMMA_F32_16X16X4_F32 (full f32 precision) so it co-executes with the
//   v_exp_f32 stream on the VALU.
// Stage 2: per-batch alpha wavefront from LDS; blank/lab staged with
//   global_load_async_to_lds_b128 (ASYNCcnt) — the CDNA5 bulk-fill path.
// Stage 3: mean of B losses.
// ---------------------------------------------------------------------------

typedef __attribute__((ext_vector_type(2))) float v2f;
typedef __attribute__((ext_vector_type(8))) float v8f;

#define NEG_SENT (-1.0e30f)

// Sum of one f32 per lane across the full wave32, using 5 f32 WMMAs.
// WMMA #1: A-VGPR0 = s (A-VGPR1 = 0), B = all-ones 4x16
//   -> D[m,n] = (sum of the 2 partials mapped to row m), constant over n,
//      i.e. each D VGPR is constant within each 16-lane half.
// WMMAs #2..#5: A = all-ones, B = consecutive VGPR pairs of D, accumulating:
//   each step adds the 4 distinct half-constants of that pair; after 4 steps
//   every element of acc equals the total sum (robust to exact lane layout).
__device__ __forceinline__ float wave_sum_wmma_f32(float s) {
  v2f a;    a[0] = s;    a[1] = 0.0f;
  v2f ones; ones[0] = 1.0f; ones[1] = 1.0f;
  v8f z   = {0.f, 0.f, 0.f, 0.f, 0.f, 0.f, 0.f, 0.f};
  v8f d1  = __builtin_amdgcn_wmma_f32_16x16x4_f32(false, a, false, ones,
                                                  (short)0, z, false, false);
  v8f acc = {0.f, 0.f, 0.f, 0.f, 0.f, 0.f, 0.f, 0.f};
  v2f bp;
  bp[0] = d1[0]; bp[1] = d1[1];
  acc = __builtin_amdgcn_wmma_f32_16x16x4_f32(false, ones, false, bp,
                                              (short)0, acc, false, false);
  bp[0] = d1[2]; bp[1] = d1[3];
  acc = __builtin_amdgcn_wmma_f32_16x16x4_f32(false, ones, false, bp,
                                              (short)0, acc, false, false);
  bp[0] = d1[4]; bp[1] = d1[5];
  acc = __builtin_amdgcn_wmma_f32_16x16x4_f32(false, ones, false, bp,
                                              (short)0, acc, false, false);
  bp[0] = d1[6]; bp[1] = d1[7];
  acc = __builtin_amdgcn_wmma_f32_16x16x4_f32(false, ones, false, bp,
                                              (short)0, acc, false, false);
  return acc[0];
}

// ------------------------- Stage 1: log-softmax gather ---------------------
// One wave per (b,t,u) cell; 8 x global_load_b128 per lane = 4KB row.
__global__ __launch_bounds__(256) void rnnt_lse_gather(
    const float* __restrict__ logits, const int* __restrict__ y,
    float* __restrict__ blank_out, float* __restrict__ lab_out,
    int T, int U, int cells) {
  const int U1   = U + 1;
  const int wave = threadIdx.x >> 5;
  const int lane = threadIdx.x & 31;
  int cell = blockIdx.x * 8 + wave;
  cell = (cell < cells) ? cell : (cells - 1);  // keep EXEC all-ones for WMMA

  int b = cell / (T * U1);
  int r = cell - b * (T * U1);
  int t = r / U1;
  int u = r - t * U1;

  const float4* src = (const float4*)(logits + (size_t)cell * 1024);
  const int yv = (u < U) ? y[b * U + u] : 0;         // wave-uniform
  const int cy = yv >> 7;                            // chunk holding label
  const int ly = (yv >> 2) & 31;                     // lane holding label
  const int jy = yv & 3;                             // float4 component

  float s = 0.f, blank_keep = 0.f, lab_keep = 0.f;
#pragma unroll
  for (int c = 0; c < 8; ++c) {
    float4 x = src[c * 32 + lane];                   // coalesced b128
    s += __expf(x.x) + __expf(x.y) + __expf(x.z) + __expf(x.w);
    // blank idx 1023 lives at chunk 7, lane 31, component .w
    blank_keep = (c == 7 && lane == 31) ? x.w : blank_keep;
    float xj = (jy == 0) ? x.x : ((jy == 1) ? x.y : ((jy == 2) ? x.z : x.w));
    lab_keep = (c == cy && lane == ly) ? xj : lab_keep;
  }

  float total = wave_sum_wmma_f32(s);                // matrix-pipe reduction
  float lse   = __logf(total);
  float blankv = __shfl(blank_keep, 31, 32);
  float labv   = __shfl(lab_keep, ly, 32);

  if (lane == 0) {
    blank_out[cell] = blankv - lse;
    if (u < U) lab_out[(b * T + t) * U + u] = labv - lse;
  }
}

// ----------------------- Stage 2: alpha wavefront --------------------------
__device__ __forceinline__ float logaddexpf_dev(float a, float b) {
  float m = fmaxf(a, b);
  return m + __logf(__expf(a - m) + __expf(b - m));
}

// Bulk global -> LDS fill via the CDNA5 async path (ASYNCcnt-tracked).
__device__ __forceinline__ void async_copy_g2l(uint32_t lds_off, float* lds_ptr,
                                               const float* gsrc,
                                               uint32_t nbytes, int tid,
                                               int nthr) {
  uint32_t main_sz = nbytes & ~15u;
  for (uint32_t ofs = (uint32_t)tid * 16u; ofs < main_sz;
       ofs += (uint32_t)nthr * 16u) {
    unsigned long long ga =
        (unsigned long long)(uintptr_t)((const char*)gsrc + ofs);
    uint32_t la = lds_off + ofs;
    asm volatile("global_load_async_to_lds_b128 %0, %1, off"
                 :: "v"(la), "v"(ga) : "memory");
  }
  // safety tail (not exercised for the given shapes; sizes are 16B multiples)
  for (uint32_t o = main_sz + (uint32_t)tid * 4u; o < nbytes;
       o += (uint32_t)nthr * 4u)
    lds_ptr[o >> 2] = gsrc[o >> 2];
}

__global__ void rnnt_alpha(const float* __restrict__ blank,
                           const float* __restrict__ lab,
                           const int* __restrict__ logit_lens,
                           const int* __restrict__ y_lens,
                           float* __restrict__ losses, int T, int U) {
  extern __shared__ float smem[];
  const int U1  = U + 1;
  const int b   = blockIdx.x;
  const int tid = threadIdx.x;
  const int nthr = blockDim.x;

  float* blankS = smem;                         // T*(U+1) floats
  float* labS   = smem + (size_t)T * U1;        // T*U floats
  float* dbuf   = labS + (size_t)T * U;         // 2*(U1+1) floats

  const float* gb = blank + (size_t)b * T * U1;
  const float* gl = lab   + (size_t)b * T * U;

  // generic LDS pointers carry the LDS byte offset in their low 32 bits
  async_copy_g2l((uint32_t)(uintptr_t)blankS, blankS, gb,
                 (uint32_t)((size_t)T * U1 * 4), tid, nthr);
  async_copy_g2l((uint32_t)(uintptr_t)labS, labS, gl,
                 (uint32_t)((size_t)T * U * 4), tid, nthr);
  __builtin_amdgcn_s_wait_asynccnt(0);
  __syncthreads();

  const int t_idx = logit_lens[b] - 1;
  const int ylen  = y_lens[b];

  float* dprev = dbuf;
  float* dcur  = dbuf + (U1 + 1);
  float  myloss = 0.f;
  const int u = tid;
  const int D = T + U;  // diagonals d = t + u, d in [0, T-1+U]

  for (int d = 0; d < D; ++d) {
    const int t = d - u;
    if (u <= U && t >= 0 && t < T) {
      float val;
      if (d == 0) {
        val = 0.f;  // alpha[0,0] = log 1
      } else {
        float up = (t >= 1) ? dprev[u] + blankS[(t - 1) * U1 + u] : NEG_SENT;
        float lf = (u >= 1) ? dprev[u - 1] + labS[t * U + (u - 1)] : NEG_SENT;
        val = logaddexpf_dev(up, lf);
      }
      if (t == t_idx && u == ylen) myloss = val;
      dcur[u] = val;
    }
    __syncthreads();
    float* tmp = dprev; dprev = dcur; dcur = tmp;
  }

  if (u == ylen) losses[b] = -(myloss + blankS[t_idx * U1 + ylen]);
}

// ----------------------------- Stage 3: mean -------------------------------
__global__ void rnnt_finalize(const float* __restrict__ losses,
                              float* __restrict__ out, int B) {
  if (threadIdx.x == 0 && blockIdx.x == 0) {
    float s = 0.f;
    for (int i = 0; i < B; ++i) s += losses[i];
    out[0] = s / (float)B;
  }
}

// ---------------------------------------------------------------------------
extern "C" void kernel_launch(void* const* d_in, const int* in_sizes, int n_in,
                              void* d_out, int out_size, void* d_ws,
                              size_t ws_size, hipStream_t stream) {
  const float* logits     = (const float*)d_in[0];  // [B,T,U+1,1024] f32
  const int*   y          = (const int*)d_in[1];    // [B,U]
  const int*   logit_lens = (const int*)d_in[2];    // [B]
  const int*   y_lens     = (const int*)d_in[3];    // [B]
  float*       out        = (float*)d_out;          // scalar

  const int B  = in_sizes[2];
  const int U  = in_sizes[1] / B;
  const int U1 = U + 1;
  const long long tot = (long long)in_sizes[0];
  const int T  = (int)(tot / ((long long)B * U1 * 1024));

  float* blank_ws = (float*)d_ws;                       // B*T*(U+1)
  float* lab_ws   = blank_ws + (size_t)B * T * U1;      // B*T*U
  float* loss_ws  = lab_ws + (size_t)B * T * U;         // B

  const int cells  = B * T * U1;                        // 133120
  const int blocks = (cells + 7) / 8;                   // 8 waves/block
  rnnt_lse_gather<<<blocks, 256, 0, stream>>>(logits, y, blank_ws, lab_ws,
                                              T, U, cells);

  const int nthr = ((U1 + 31) / 32) * 32;               // 96 threads (3 waves)
  const size_t shmem =
      ((size_t)T * U1 + (size_t)T * U + 2 * (size_t)(U1 + 1)) * sizeof(float);
  rnnt_alpha<<<B, nthr, shmem, stream>>>(blank_ws, lab_ws, logit_lens, y_lens,
                                         loss_ws, T, U);

  rnnt_finalize<<<1, 32, 0, stream>>>(loss_ws, out, B);
}